// CrossGraphSample_17824114278454
// MI455X (gfx1250) — compile-verified
//
#include <hip/hip_runtime.h>
#include <hip/hip_bf16.h>

#define B_    8
#define C_    256
#define N_    2048
#define M_    2048
#define KTOP  1638          // round(2048 * 0.8)
#define ROWS  16            // adj rows per workgroup (LDS strip = 128KB)
#define MT    (M_ / ROWS)   // 128 m-tiles
#define NWG   (B_ * MT)     // 1024 fused workgroups

typedef __attribute__((ext_vector_type(2))) float v2f;
typedef __attribute__((ext_vector_type(8))) float v8f;

// ---------------------------------------------------------------------------
// prep: per-(b,n) L2 norm over C; emit transposed raw (xt) and normalized (xnt)
// layouts [B][N][C] so the WMMA K-dim (C) is contiguous.
// ---------------------------------------------------------------------------
__global__ __launch_bounds__(256) void prep_x(const float* __restrict__ x,
                                              float* __restrict__ xt,
                                              float* __restrict__ xnt) {
    const int b = blockIdx.y, n = blockIdx.x, c = threadIdx.x;
    float v = x[((size_t)b * C_ + c) * N_ + n];
    __shared__ float red[256];
    red[c] = v * v;
    __syncthreads();
    for (int s = 128; s > 0; s >>= 1) {
        if (c < s) red[c] += red[c + s];
        __syncthreads();
    }
    const float rinv = 1.0f / fmaxf(sqrtf(red[0]), 1e-12f);
    const size_t o = ((size_t)b * N_ + n) * C_ + c;
    xt[o]  = v;
    xnt[o] = v * rinv;
}

__global__ __launch_bounds__(256) void prep_t(const float* __restrict__ tg,
                                              float* __restrict__ tgn) {
    const int b = blockIdx.y, m = blockIdx.x, c = threadIdx.x;
    float v = tg[((size_t)b * C_ + c) * M_ + m];
    __shared__ float red[256];
    red[c] = v * v;
    __syncthreads();
    for (int s = 128; s > 0; s >>= 1) {
        if (c < s) red[c] += red[c + s];
        __syncthreads();
    }
    const float rinv = 1.0f / fmaxf(sqrtf(red[0]), 1e-12f);
    tgn[((size_t)b * M_ + m) * C_ + c] = v * rinv;
}

// ---------------------------------------------------------------------------
// fused: GEMM1 (f32 WMMA) -> softmax -> exact top-k mask (radix bisection on
// float bit patterns) -> GEMM2 (f32 WMMA) -> LeakyReLU -> preBN + BN partials.
// One workgroup = one (b, 16-row m-strip); adj strip lives entirely in LDS.
// ---------------------------------------------------------------------------
__global__ __launch_bounds__(256) void fused_kernel(
    const float* __restrict__ xt,   // [B][N][C] raw transposed
    const float* __restrict__ xnt,  // [B][N][C] normalized transposed
    const float* __restrict__ tgn,  // [B][M][C] normalized transposed
    float* __restrict__ preBN,      // [B][M][C] post-LeakyReLU pre-BN
    float* __restrict__ psum,       // [NWG][C]  per-WG channel sums
    float* __restrict__ psq)        // [NWG][C]  per-WG channel sum-of-squares
{
    extern __shared__ float smem[];
    float* adjS = smem;                 // ROWS * N_ floats (128KB)
    float* As   = smem + ROWS * N_;     // ROWS * C_ floats (16KB)

    const int b    = blockIdx.y;
    const int m0   = blockIdx.x * ROWS;
    const int tid  = threadIdx.x;
    const int lane = tid & 31;
    const int wave = tid >> 5;
    const int lmod = lane & 15;
    const int hsel = lane >> 4;         // lane half select (0/1)

    // stage A = tgn rows [m0 .. m0+15][0..C) into LDS
    for (int i = tid; i < ROWS * C_; i += 256)
        As[i] = tgn[((size_t)b * M_ + m0 + (i >> 8)) * C_ + (i & 255)];
    __syncthreads();

    // ---- Phase 1: adj[16][2048] = A(16xC) x Bn(Cx2048), K=C in steps of 4 ----
    for (int t = 0; t < 16; ++t) {
        const int n0 = (wave * 16 + t) * 16;
        const float* xr = xnt + ((size_t)b * N_ + n0 + lmod) * C_;
        v8f acc = {};
#pragma unroll 8
        for (int k0 = 0; k0 < C_; k0 += 4) {
            v2f a, bb;
            a.x  = As[lmod * C_ + k0 + 2 * hsel];        // A[m=lmod][k, k+1 | k+2, k+3]
            a.y  = As[lmod * C_ + k0 + 2 * hsel + 1];
            bb.x = xr[k0 + 2 * hsel];                    // B[k][n=n0+lmod]
            bb.y = xr[k0 + 2 * hsel + 1];
            acc = __builtin_amdgcn_wmma_f32_16x16x4_f32(
                false, a, false, bb, (short)0, acc, false, false);
        }
#pragma unroll
        for (int r = 0; r < 8; ++r)                      // D: VGPR r -> m = r + 8*hsel
            adjS[(r + 8 * hsel) * N_ + n0 + lmod] = acc[r];
    }
    __syncthreads();

    // ---- Phase 2+3: per-row softmax + top-k threshold mask (2 rows / wave) ----
    for (int rr = 0; rr < 2; ++rr) {
        const int row = wave * 2 + rr;
        float* rowp = adjS + row * N_;

        float mx = -3.402823466e38f;
        for (int i = lane; i < N_; i += 32) mx = fmaxf(mx, rowp[i]);
        for (int m = 16; m >= 1; m >>= 1) mx = fmaxf(mx, __shfl_xor(mx, m, 32));

        float sum = 0.0f;
        for (int i = lane; i < N_; i += 32) {
            float e = __expf(rowp[i] - mx);
            rowp[i] = e;
            sum += e;
        }
        for (int m = 16; m >= 1; m >>= 1) sum += __shfl_xor(sum, m, 32);
        const float inv = 1.0f / sum;
        for (int i = lane; i < N_; i += 32) rowp[i] *= inv;

        // k-th largest via binary search on positive-float bit order
        unsigned lo = 0u, hi = 0x3F800001u;  // softmax values in [0,1]
        while (hi - lo > 1u) {
            const unsigned mid = (lo + hi) >> 1;
            const float t = __uint_as_float(mid);
            int cnt = 0;
            for (int i = lane; i < N_; i += 32) cnt += (rowp[i] >= t) ? 1 : 0;
            for (int m = 16; m >= 1; m >>= 1) cnt += __shfl_xor(cnt, m, 32);
            if (cnt >= KTOP) lo = mid; else hi = mid;
        }
        const float thr = __uint_as_float(lo);
        for (int i = lane; i < N_; i += 32) {
            float v = rowp[i];
            rowp[i] = (v >= thr) ? v : 0.0f;
        }
    }
    __syncthreads();

    // ---- Phase 4: out(16m x C) = adj(16x2048) x xtT(2048xC), K=N in steps of 4
    for (int ct = 0; ct < 2; ++ct) {
        const int c0 = (wave * 2 + ct) * 16;   // waves own disjoint 32-ch slices
        v8f acc = {};
#pragma unroll 4
        for (int k0 = 0; k0 < N_; k0 += 4) {
            v2f a, bb;
            a.x = adjS[lmod * N_ + k0 + 2 * hsel];
            a.y = adjS[lmod * N_ + k0 + 2 * hsel + 1];
            const float* xr0 = xt + ((size_t)b * N_ + k0 + 2 * hsel) * C_ + c0 + lmod;
            bb.x = xr0[0];
            bb.y = xr0[C_];
            acc = __builtin_amdgcn_wmma_f32_16x16x4_f32(
                false, a, false, bb, (short)0, acc, false, false);
        }
        float s = 0.0f, q = 0.0f;
#pragma unroll
        for (int r = 0; r < 8; ++r) {
            float v = acc[r];
            v = (v >= 0.0f) ? v : 0.01f * v;   // LeakyReLU
            acc[r] = v;
            s += v;
            q += v * v;
        }
#pragma unroll
        for (int r = 0; r < 8; ++r)
            preBN[((size_t)b * M_ + m0 + r + 8 * hsel) * C_ + c0 + lmod] = acc[r];
        // combine lane halves (same channel, disjoint m) -> 16-row channel sums
        s += __shfl_xor(s, 16, 32);
        q += __shfl_xor(q, 16, 32);
        if (lane < 16) {
            const int wg = blockIdx.y * MT + blockIdx.x;
            psum[(size_t)wg * C_ + c0 + lane] = s;
            psq [(size_t)wg * C_ + c0 + lane] = q;
        }
    }
}

// ---------------------------------------------------------------------------
// BN stats: one block per channel; deterministic fixed-order double reduction.
// stats[c] = {scale, shift} folding mean/var/weight/bias into one FMA.
// ---------------------------------------------------------------------------
__global__ __launch_bounds__(256) void bn_stats(const float* __restrict__ psum,
                                                const float* __restrict__ psq,
                                                const float* __restrict__ bnw,
                                                const float* __restrict__ bnb,
                                                float* __restrict__ stats) {
    const int c = blockIdx.x, t = threadIdx.x;
    double s = 0.0, q = 0.0;
    for (int p = t; p < NWG; p += 256) {
        s += (double)psum[(size_t)p * C_ + c];
        q += (double)psq [(size_t)p * C_ + c];
    }
    __shared__ double rs[256], rq[256];
    rs[t] = s; rq[t] = q;
    __syncthreads();
    for (int st = 128; st > 0; st >>= 1) {
        if (t < st) { rs[t] += rs[t + st]; rq[t] += rq[t + st]; }
        __syncthreads();
    }
    if (t == 0) {
        const double cnt  = (double)B_ * (double)M_;
        const double mean = rs[0] / cnt;
        const double var  = rq[0] / cnt - mean * mean;
        const double scale = (double)bnw[c] / sqrt(var + 1e-5);
        stats[2 * c]     = (float)scale;
        stats[2 * c + 1] = (float)((double)bnb[c] - mean * scale);
    }
}

// ---------------------------------------------------------------------------
// finalize: LDS-tiled transpose [B][M][C] -> [B][C][M] with BN affine,
// gamma scale and residual add of target_g. Coalesced on both sides.
// ---------------------------------------------------------------------------
__global__ __launch_bounds__(256) void finalize(const float* __restrict__ preBN,
                                                const float* __restrict__ stats,
                                                const float* __restrict__ gamma,
                                                const float* __restrict__ tg,
                                                float* __restrict__ out) {
    __shared__ float tile[32][33];
    const int b  = blockIdx.z;
    const int m0 = blockIdx.x * 32, c0 = blockIdx.y * 32;
    const int tx = threadIdx.x, ty = threadIdx.y;   // block (32, 8)
#pragma unroll
    for (int i = 0; i < 32; i += 8)
        tile[ty + i][tx] = preBN[((size_t)b * M_ + m0 + ty + i) * C_ + c0 + tx];
    __syncthreads();
    const float g = gamma[0];
#pragma unroll
    for (int i = 0; i < 32; i += 8) {
        const int c = c0 + ty + i, m = m0 + tx;
        float v = tile[tx][ty + i];
        v = v * stats[2 * c] + stats[2 * c + 1];
        const size_t o = ((size_t)b * C_ + c) * M_ + m;
        out[o] = v * g + tg[o];
    }
}

// ---------------------------------------------------------------------------
extern "C" void kernel_launch(void* const* d_in, const int* in_sizes, int n_in,
                              void* d_out, int out_size, void* d_ws, size_t ws_size,
                              hipStream_t stream) {
    const float* x     = (const float*)d_in[0];  // [B][C][N]
    const float* tg    = (const float*)d_in[1];  // [B][C][M]
    const float* gamma = (const float*)d_in[2];  // [1]
    const float* bnw   = (const float*)d_in[3];  // [C]
    const float* bnb   = (const float*)d_in[4];  // [C]
    float* out = (float*)d_out;                  // [B][C][M]

    float* ws    = (float*)d_ws;
    float* xt    = ws;                                    // B*N*C
    float* xnt   = xt    + (size_t)B_ * N_ * C_;          // B*N*C
    float* tgn   = xnt   + (size_t)B_ * N_ * C_;          // B*M*C
    float* preBN = tgn   + (size_t)B_ * M_ * C_;          // B*M*C
    float* psum  = preBN + (size_t)B_ * M_ * C_;          // NWG*C
    float* psq   = psum  + (size_t)NWG * C_;              // NWG*C
    float* stats = psq   + (size_t)NWG * C_;              // 2*C

    prep_x<<<dim3(N_, B_), 256, 0, stream>>>(x, xt, xnt);
    prep_t<<<dim3(M_, B_), 256, 0, stream>>>(tg, tgn);

    const size_t smem = (size_t)(ROWS * N_ + ROWS * C_) * sizeof(float); // 144KB
    fused_kernel<<<dim3(MT, B_), 256, smem, stream>>>(xt, xnt, tgn, preBN, psum, psq);

    bn_stats<<<C_, 256, 0, stream>>>(psum, psq, bnw, bnb, stats);
    finalize<<<dim3(M_ / 32, C_ / 32, B_), dim3(32, 8), 0, stream>>>(
        preBN, stats, gamma, tg, out);
}